// ConvSE3_22290880266593
// MI455X (gfx1250) — compile-verified
//
#include <hip/hip_runtime.h>

typedef float v2f __attribute__((ext_vector_type(2)));
typedef float v8f __attribute__((ext_vector_type(8)));

#define NNODES 256
#define KNB    32
#define NEDGE  (NNODES*KNB)   /* 8192 */
#define CCH    32
#define MIDD   128

struct MlpPtrs {
  const float *w1,*b1,*g1,*be1,*w2,*b2,*g2,*be2,*w3,*b3;
};

__device__ __forceinline__ float wave_allreduce_add(float v) {
#pragma unroll
  for (int off = 16; off >= 1; off >>= 1) v += __shfl_xor(v, off, 32);
  return v;
}
// reduce within each 16-lane half (xor masks < 16 never cross halves)
__device__ __forceinline__ float half_allreduce_add(float v) {
#pragma unroll
  for (int off = 8; off >= 1; off >>= 1) v += __shfl_xor(v, off, 32);
  return v;
}

// ---------------------------------------------------------------------------
// K0: one-time transpose of the four W3 matrices (128 x F, row-major) into
//     W3T (F x 128, row-major) so K3's A fragments become contiguous b64
//     loads.  Reads are fully coalesced along m; total data is only 3 MB.
//     W3T pair offsets: pair p<3 -> p*131072, pair 3 -> 393216.
// ---------------------------------------------------------------------------
__global__ __launch_bounds__(256) void k0_w3t(
    MlpPtrs m0, MlpPtrs m1, MlpPtrs m2, MlpPtrs m3, float* __restrict__ W3T)
{
  int pair = blockIdx.y;
  const MlpPtrs P = (pair==0) ? m0 : (pair==1) ? m1 : (pair==2) ? m2 : m3;
  const int F = (pair==3) ? 3072 : 1024;
  size_t off = (pair==3) ? (size_t)3*131072 : (size_t)pair*131072;
  size_t total = (size_t)F * MIDD;
  for (size_t idx = (size_t)blockIdx.x*256 + threadIdx.x; idx < total;
       idx += (size_t)gridDim.x*256) {
    int k    = (int)(idx / F);
    int mcol = (int)(idx % F);
    W3T[off + (size_t)mcol*MIDD + k] = P.w3[idx];
  }
}

// ---------------------------------------------------------------------------
// K1: per-edge prep.  One wave (32 lanes) per edge.
//   - h1 = relu(LN(d*w1+b1)) for all 4 MLPs -> H1[mlp][edge][128]
//   - gather neighbor features, precontract with basis:
//       g0[e][i], v00[e][i]=B00*g0, v10[e][i]=sum_q B10[q]*g1[i][q],
//       u11[e][(i*3+f)*3+p] = sum_q B11[p][q][f]*g1[i][q]
// ---------------------------------------------------------------------------
__device__ __forceinline__ void k1_mlp_h1(const MlpPtrs& P, int e, int lane,
                                          float d, float* __restrict__ H1m) {
  float t[4]; float s1 = 0.f;
#pragma unroll
  for (int j = 0; j < 4; ++j) {
    int dim = lane + 32*j;
    t[j] = fmaf(d, P.w1[dim], P.b1[dim]);
    s1 += t[j];
  }
  s1 = wave_allreduce_add(s1);
  float mu = s1 * (1.f/128.f);
  float s2 = 0.f;
#pragma unroll
  for (int j = 0; j < 4; ++j) { float c = t[j]-mu; s2 = fmaf(c, c, s2); }
  s2 = wave_allreduce_add(s2);
  float rinv = rsqrtf(s2*(1.f/128.f) + 1e-5f);
#pragma unroll
  for (int j = 0; j < 4; ++j) {
    int dim = lane + 32*j;
    float h = fmaf((t[j]-mu)*rinv, P.g1[dim], P.be1[dim]);
    H1m[(size_t)e*MIDD + dim] = fmaxf(h, 0.f);
  }
}

__global__ __launch_bounds__(256) void k1_prep(
    const float* __restrict__ rel, const int* __restrict__ nidx,
    const float* __restrict__ x0, const float* __restrict__ x1,
    const float* __restrict__ b00, const float* __restrict__ b10,
    const float* __restrict__ b11,
    MlpPtrs m0, MlpPtrs m1, MlpPtrs m2, MlpPtrs m3,
    float* __restrict__ H1, float* __restrict__ g0w, float* __restrict__ v00w,
    float* __restrict__ v10w, float* __restrict__ u11w)
{
  int e    = blockIdx.x * (blockDim.x >> 5) + (threadIdx.x >> 5);
  int lane = threadIdx.x & 31;
  float d = rel[e];
  k1_mlp_h1(m0, e, lane, d, H1 + (size_t)0*NEDGE*MIDD);
  k1_mlp_h1(m1, e, lane, d, H1 + (size_t)1*NEDGE*MIDD);
  k1_mlp_h1(m2, e, lane, d, H1 + (size_t)2*NEDGE*MIDD);
  k1_mlp_h1(m3, e, lane, d, H1 + (size_t)3*NEDGE*MIDD);

  int nb = nidx[e];
  int i  = lane;
  float gg0 = x0[nb*CCH + i];
  float gq0 = x1[(nb*CCH + i)*3 + 0];
  float gq1 = x1[(nb*CCH + i)*3 + 1];
  float gq2 = x1[(nb*CCH + i)*3 + 2];
  g0w[e*CCH + i]  = gg0;
  v00w[e*CCH + i] = b00[e] * gg0;
  v10w[e*CCH + i] = b10[e*3+0]*gq0 + b10[e*3+1]*gq1 + b10[e*3+2]*gq2;
#pragma unroll
  for (int f = 0; f < 3; ++f) {
#pragma unroll
    for (int p = 0; p < 3; ++p) {
      float u = b11[((e*3+p)*3+0)*3+f]*gq0
              + b11[((e*3+p)*3+1)*3+f]*gq1
              + b11[((e*3+p)*3+2)*3+f]*gq2;
      u11w[(size_t)e*288 + (i*3+f)*3 + p] = u;
    }
  }
}

// ---------------------------------------------------------------------------
// K2: H2 = relu(LN(H1 @ W2 + b2)) per MLP, via V_WMMA_F32_16X16X4_F32.
//   grid = (64, 4 mlps), block = 256 (8 waves).  Wave owns a 16-edge M-tile,
//   all 8 N-tiles (N=128), K=128 in 32 wmma steps.  W2 staged in LDS.
//   LayerNorm done in registers with 16-lane shfl reductions on D fragments.
// Fragment layouts (ISA 7.12.2, 16x16x4 f32):
//   A: lane L holds A[L%16][(L/16)*2 + j], j=0,1
//   B: lane L holds B[(L/16)*2 + j][L%16]
//   D: lane L, vgpr v holds D[v + 8*(L/16)][L%16]
// ---------------------------------------------------------------------------
__global__ __launch_bounds__(256) void k2_gemm_h2(
    const float* __restrict__ H1,
    MlpPtrs m0, MlpPtrs m1, MlpPtrs m2, MlpPtrs m3,
    float* __restrict__ H2)
{
  __shared__ float w2s[MIDD*MIDD];   // 64 KB
  int mlp = blockIdx.y;
  const MlpPtrs P = (mlp==0) ? m0 : (mlp==1) ? m1 : (mlp==2) ? m2 : m3;

  for (int t = threadIdx.x; t < MIDD*MIDD; t += 256) w2s[t] = P.w2[t];
  __syncthreads();

  int wave = threadIdx.x >> 5, lane = threadIdx.x & 31;
  int half = lane >> 4,       lm   = lane & 15;
  int eb = (blockIdx.x*8 + wave)*16;

  const float* Abase = H1 + ((size_t)mlp*NEDGE + (eb + lm))*MIDD + half*2;

  v8f D[8];
#pragma unroll
  for (int nt = 0; nt < 8; ++nt) D[nt] = (v8f){0.f,0.f,0.f,0.f,0.f,0.f,0.f,0.f};

  for (int ks = 0; ks < 32; ++ks) {
    v2f a = *(const v2f*)(Abase + ks*4);
    int kb = ks*4 + half*2;
#pragma unroll
    for (int nt = 0; nt < 8; ++nt) {
      v2f b;
      b[0] = w2s[(kb+0)*MIDD + nt*16 + lm];
      b[1] = w2s[(kb+1)*MIDD + nt*16 + lm];
      D[nt] = __builtin_amdgcn_wmma_f32_16x16x4_f32(
                  false, a, false, b, (short)0, D[nt], false, false);
    }
  }

  float bcol[8], gcol[8], becol[8];
#pragma unroll
  for (int nt = 0; nt < 8; ++nt) {
    int col = nt*16 + lm;
    bcol[nt] = P.b2[col]; gcol[nt] = P.g2[col]; becol[nt] = P.be2[col];
  }
#pragma unroll
  for (int v = 0; v < 8; ++v) {
    float xv[8]; float s1 = 0.f;
#pragma unroll
    for (int nt = 0; nt < 8; ++nt) { xv[nt] = D[nt][v] + bcol[nt]; s1 += xv[nt]; }
    s1 = half_allreduce_add(s1);            // row lives in one 16-lane half
    float mu = s1 * (1.f/128.f);
    float s2 = 0.f;
#pragma unroll
    for (int nt = 0; nt < 8; ++nt) { float c = xv[nt]-mu; s2 = fmaf(c, c, s2); }
    s2 = half_allreduce_add(s2);
    float rinv = rsqrtf(s2*(1.f/128.f) + 1e-5f);
    int edge = eb + v + 8*half;
    float* outp = H2 + ((size_t)mlp*NEDGE + edge)*MIDD;
#pragma unroll
    for (int nt = 0; nt < 8; ++nt) {
      float h = fmaf((xv[nt]-mu)*rinv, gcol[nt], becol[nt]);
      outp[nt*16 + lm] = fmaxf(h, 0.f);
    }
  }
}

// ---------------------------------------------------------------------------
// K3: fused  Y^T = W3^T @ H2^T  (+b3)  with immediate basis contraction.
//   Transposed so lane = edge: lane's 8 D values/tile are FMA'd against its
//   edge's precontracted vectors.  Tile order is o-major -> accumulators are
//   compile-time indexed (1 or 3 regs per edge-tile).
//   A fragments come from the pre-transposed W3T -> single b64 loads.
//   grid = (32, 4 pairs, 2 o-halves), block = 256 (8 waves), wave owns 32
//   edges, 2x2 tile blocking (2 M-tiles x 2 edge-tiles) for fragment reuse.
//   pair: 0=(di0,do0) vec v00 ; 1=(di0,do1) vec g0 -> s01[o] ;
//         2=(di1,do0) vec v10 ; 3=(di1,do1) u11 -> acc11[o][p]
// ---------------------------------------------------------------------------
__global__ __launch_bounds__(256) void k3_pairs(
    const float* __restrict__ H2, const float* __restrict__ W3T,
    MlpPtrs m0, MlpPtrs m1, MlpPtrs m2, MlpPtrs m3,
    const float* __restrict__ g0w,  const float* __restrict__ v00w,
    const float* __restrict__ v10w, const float* __restrict__ u11w,
    float* __restrict__ acc00, float* __restrict__ s01w,
    float* __restrict__ acc10, float* __restrict__ acc11)
{
  int pair = blockIdx.y;
  const MlpPtrs P = (pair==0) ? m0 : (pair==1) ? m1 : (pair==2) ? m2 : m3;
  const int CW  = (pair==3) ? 96 : 32;       // columns per output channel o
  const int MT2 = (pair==3) ? 3 : 1;         // m-tile *pairs* per o

  const float* cvec = (pair==0) ? v00w : (pair==1) ? g0w : v10w;
  float* outw = (pair==0) ? acc00 : (pair==1) ? s01w : (pair==2) ? acc10 : acc11;
  const float* W3Tp = W3T + ((pair==3) ? (size_t)3*131072 : (size_t)pair*131072);
  const float* b3 = P.b3;

  int wave = threadIdx.x >> 5, lane = threadIdx.x & 31;
  int half = lane >> 4,       lm   = lane & 15;
  int eb = (blockIdx.x*8 + wave)*32;

  const float* H2p = H2 + (size_t)pair*NEDGE*MIDD;
  int e0 = eb + lm, e1 = eb + 16 + lm;
  const float* B0base = H2p + (size_t)e0*MIDD + half*2;
  const float* B1base = H2p + (size_t)e1*MIDD + half*2;

  int obeg = blockIdx.z * 16, oend = obeg + 16;
  for (int o = obeg; o < oend; ++o) {
    float acc[2][3] = {{0.f,0.f,0.f},{0.f,0.f,0.f}};   // [edge-tile][p]
    for (int mt2 = 0; mt2 < MT2; ++mt2) {
      int mcol0 = o*CW + mt2*32;
      const float* A0base = W3Tp + (size_t)(mcol0      + lm)*MIDD + half*2;
      const float* A1base = W3Tp + (size_t)(mcol0 + 16 + lm)*MIDD + half*2;
      v8f Dt[2][2];
#pragma unroll
      for (int mi = 0; mi < 2; ++mi)
#pragma unroll
        for (int ei = 0; ei < 2; ++ei)
          Dt[mi][ei] = (v8f){0.f,0.f,0.f,0.f,0.f,0.f,0.f,0.f};

      for (int ks = 0; ks < 32; ++ks) {
        v2f a0 = *(const v2f*)(A0base + ks*4);
        v2f a1 = *(const v2f*)(A1base + ks*4);
        v2f b0 = *(const v2f*)(B0base + ks*4);
        v2f b1 = *(const v2f*)(B1base + ks*4);
        Dt[0][0] = __builtin_amdgcn_wmma_f32_16x16x4_f32(false,a0,false,b0,(short)0,Dt[0][0],false,false);
        Dt[0][1] = __builtin_amdgcn_wmma_f32_16x16x4_f32(false,a0,false,b1,(short)0,Dt[0][1],false,false);
        Dt[1][0] = __builtin_amdgcn_wmma_f32_16x16x4_f32(false,a1,false,b0,(short)0,Dt[1][0],false,false);
        Dt[1][1] = __builtin_amdgcn_wmma_f32_16x16x4_f32(false,a1,false,b1,(short)0,Dt[1][1],false,false);
      }
      // contraction: lane's edge is fixed per edge-tile; cols = rows of Y^T
#pragma unroll
      for (int mi = 0; mi < 2; ++mi) {
#pragma unroll
        for (int ei = 0; ei < 2; ++ei) {
          int edge = eb + ei*16 + lm;
          int colb = mcol0 + mi*16 + 8*half;
#pragma unroll
          for (int v = 0; v < 8; ++v) {
            int col = colb + v;
            float y = Dt[mi][ei][v] + b3[col];
            int loc = col - o*CW;                 // i (pairs 0-2) or i*3+f (pair 3)
            if (pair == 3) {
              const float* u = u11w + (size_t)edge*288 + loc*3;
              acc[ei][0] = fmaf(y, u[0], acc[ei][0]);
              acc[ei][1] = fmaf(y, u[1], acc[ei][1]);
              acc[ei][2] = fmaf(y, u[2], acc[ei][2]);
            } else {
              acc[ei][0] = fmaf(y, cvec[edge*CCH + loc], acc[ei][0]);
            }
          }
        }
      }
    }
    // finalize channel o: combine the two lane halves (same edge set)
#pragma unroll
    for (int ei = 0; ei < 2; ++ei) {
      int edge = eb + ei*16 + lm;
      if (pair == 3) {
#pragma unroll
        for (int p = 0; p < 3; ++p) {
          float r = acc[ei][p];
          r += __shfl_xor(r, 16, 32);
          if (half == 0) outw[(size_t)edge*96 + o*3 + p] = r;
        }
      } else {
        float r = acc[ei][0];
        r += __shfl_xor(r, 16, 32);
        if (half == 0) outw[(size_t)edge*CCH + o] = r;
      }
    }
  }
}

// ---------------------------------------------------------------------------
// K4: masked mean over neighbors + self interaction + residual.
//   block per node (256 blocks x 128 threads).
//   acc1_full[e][o][p] = acc11 + B01[e][p]*s01[e][o]
// ---------------------------------------------------------------------------
__global__ __launch_bounds__(128) void k4_final(
    const float* __restrict__ x0, const float* __restrict__ x1,
    const unsigned char* __restrict__ mask, const float* __restrict__ b01,
    const float* __restrict__ si0, const float* __restrict__ si1,
    const float* __restrict__ acc00, const float* __restrict__ s01w,
    const float* __restrict__ acc10, const float* __restrict__ acc11,
    float* __restrict__ out)
{
  int n = blockIdx.x;
  int t = threadIdx.x;
  __shared__ float msh[KNB];
  __shared__ float denom_s;
  __shared__ int   mtot_s;
  if (t < KNB) msh[t] = (mask[n*KNB + t] != 0) ? 1.f : 0.f;
  __syncthreads();
  if (t == 0) {
    float s = 0.f;
    for (int k = 0; k < KNB; ++k) s += msh[k];
    mtot_s  = (int)s;
    denom_s = fmaxf(s, 1.f);
  }
  __syncthreads();
  float denom = denom_s;
  int   mtot  = mtot_s;

  if (t < CCH) {                       // degree-0 output, channel d = t
    int d = t;
    float pooled = 0.f;
    for (int k = 0; k < KNB; ++k) {
      int e = n*KNB + k;
      pooled += msh[k] * (acc00[(size_t)e*CCH + d] + acc10[(size_t)e*CCH + d]);
    }
    pooled /= denom;
    if (mtot == 0) pooled = 0.f;
    float si = 0.f;
    for (int c = 0; c < CCH; ++c) si = fmaf(x0[n*CCH + c], si0[c*CCH + d], si);
    out[n*CCH + d] = pooled + si;
  }
  if (t < CCH*3) {                     // degree-1 output, (d,p)
    int d = t / 3, p = t % 3;
    float pooled = 0.f;
    for (int k = 0; k < KNB; ++k) {
      int e = n*KNB + k;
      float a = acc11[(size_t)e*96 + d*3 + p] + b01[e*3 + p] * s01w[(size_t)e*CCH + d];
      pooled += msh[k] * a;
    }
    pooled /= denom;
    if (mtot == 0) pooled = 0.f;
    float si = 0.f;
    for (int c = 0; c < CCH; ++c) si = fmaf(x1[(n*CCH + c)*3 + p], si1[c*CCH + d], si);
    out[NEDGE + (n*CCH + d)*3 + p] = pooled + si;
  }
}

// ---------------------------------------------------------------------------
// Host launcher
// ---------------------------------------------------------------------------
extern "C" void kernel_launch(void* const* d_in, const int* in_sizes, int n_in,
                              void* d_out, int out_size, void* d_ws, size_t ws_size,
                              hipStream_t stream) {
  const float* x0   = (const float*)d_in[0];
  const float* x1   = (const float*)d_in[1];
  const float* rel  = (const float*)d_in[2];
  const int*   nidx = (const int*)d_in[3];
  const unsigned char* msk = (const unsigned char*)d_in[4];
  const float* b00  = (const float*)d_in[5];
  const float* b01  = (const float*)d_in[6];
  const float* b10  = (const float*)d_in[7];
  const float* b11  = (const float*)d_in[8];

  MlpPtrs m[4];
  for (int i = 0; i < 4; ++i) {
    int b = 9 + i*10;
    const float* L[10];
    for (int j = 0; j < 10; ++j) L[j] = (const float*)d_in[b + j];
    if (in_sizes[b + 2] == MIDD) {
      // insertion order: w1,b1,g1,be1,w2,b2,g2,be2,w3,b3
      m[i] = { L[0], L[1], L[2], L[3], L[4], L[5], L[6], L[7], L[8], L[9] };
    } else {
      // sorted order: b1,b2,b3,be1,be2,g1,g2,w1,w2,w3
      m[i].b1 = L[0]; m[i].b2 = L[1]; m[i].b3 = L[2]; m[i].be1 = L[3];
      m[i].be2 = L[4]; m[i].g1 = L[5]; m[i].g2 = L[6];
      m[i].w1 = L[7]; m[i].w2 = L[8]; m[i].w3 = L[9];
    }
  }
  const float* si0 = (const float*)d_in[49];
  const float* si1 = (const float*)d_in[50];

  // workspace layout (floats); total ~13.9M floats = ~55.6 MB
  float* ws = (float*)d_ws;
  const size_t H1sz = (size_t)4*NEDGE*MIDD;        // 4,194,304
  float* H1    = ws;
  float* H2    = H1    + H1sz;                     // +4,194,304
  float* g0w   = H2    + H1sz;
  float* v00w  = g0w   + (size_t)NEDGE*CCH;
  float* v10w  = v00w  + (size_t)NEDGE*CCH;
  float* u11w  = v10w  + (size_t)NEDGE*CCH;        // +2,359,296
  float* acc00 = u11w  + (size_t)NEDGE*288;
  float* s01w  = acc00 + (size_t)NEDGE*CCH;
  float* acc10 = s01w  + (size_t)NEDGE*CCH;
  float* acc11 = acc10 + (size_t)NEDGE*CCH;        // +786,432
  float* W3T   = acc11 + (size_t)NEDGE*96;         // +786,432 (3 MB)

  float* out = (float*)d_out;

  dim3 g0(512, 4);
  k0_w3t<<<g0, 256, 0, stream>>>(m[0], m[1], m[2], m[3], W3T);
  k1_prep<<<NEDGE/8, 256, 0, stream>>>(rel, nidx, x0, x1, b00, b10, b11,
                                       m[0], m[1], m[2], m[3],
                                       H1, g0w, v00w, v10w, u11w);
  dim3 g2(64, 4);
  k2_gemm_h2<<<g2, 256, 0, stream>>>(H1, m[0], m[1], m[2], m[3], H2);
  dim3 g3(32, 4, 2);
  k3_pairs<<<g3, 256, 0, stream>>>(H2, W3T, m[0], m[1], m[2], m[3],
                                   g0w, v00w, v10w, u11w,
                                   acc00, s01w, acc10, acc11);
  k4_final<<<NNODES, 128, 0, stream>>>(x0, x1, msk, b01, si0, si1,
                                       acc00, s01w, acc10, acc11, out);
}